// ChebNet_46961172414540
// MI455X (gfx1250) — compile-verified
//
#include <hip/hip_runtime.h>
#include <hip/hip_bf16.h>

typedef __attribute__((ext_vector_type(2))) float v2f;
typedef __attribute__((ext_vector_type(8))) float v8f;

#define NNODES 100000
#define NEDGES 1600000
#define NGRAPH 32
#define D 64
#define KD 192            // K * D total reduction dim
#define NPAIR (KD / 2)    // 96 K-pairs
#define PSTRIDE 160       // dwords per pair-row: 128 data + 32 pad (bank-conflict free)

// ---------------- utility ----------------
__global__ void zero_kernel(float* __restrict__ p, int n) {
    int i = blockIdx.x * blockDim.x + threadIdx.x;
    if (i < n) p[i] = 0.0f;
}

// deg accumulation: deg[dst[e]] += 1
__global__ void degree_kernel(const int* __restrict__ dst, float* __restrict__ deg, int E) {
    int e = blockIdx.x * blockDim.x + threadIdx.x;
    if (e < E) atomicAdd(&deg[dst[e]], 1.0f);
}

// dinv = clamp(deg,1)^-0.5 (in place on deg buffer)
__global__ void dinv_kernel(float* __restrict__ d, int n) {
    int i = blockIdx.x * blockDim.x + threadIdx.x;
    if (i < n) d[i] = 1.0f / sqrtf(fmaxf(d[i], 1.0f));
}

// ---------------- SpMM: agg[dst] += (x[src] * dinv[src]) ----------------
// one thread = one edge * 4 features (float4 gather, 4 atomic adds into L2)
__global__ void spmm_scatter(const float* __restrict__ x,
                             const int* __restrict__ src,
                             const int* __restrict__ dst,
                             const float* __restrict__ dinv,
                             float* __restrict__ agg, int E) {
    int idx = blockIdx.x * blockDim.x + threadIdx.x;
    if (idx >= E * 16) return;
    int e  = idx >> 4;
    int f4 = (idx & 15) << 2;
    int s = __ldg(src + e);
    int d = __ldg(dst + e);
    float sc = __ldg(dinv + s);
    float4 xv = *reinterpret_cast<const float4*>(x + (size_t)s * D + f4);
    float* o = agg + (size_t)d * D + f4;
    atomicAdd(o + 0, xv.x * sc);
    atomicAdd(o + 1, xv.y * sc);
    atomicAdd(o + 2, xv.z * sc);
    atomicAdd(o + 3, xv.w * sc);
}

// T1 = -dinv * agg   (in place)
__global__ void finish_T1(float* __restrict__ X1, const float* __restrict__ dinv, int n) {
    int i = blockIdx.x * blockDim.x + threadIdx.x;
    if (i < n) X1[i] = -dinv[i >> 6] * X1[i];
}

// T2 = -2*dinv*agg - X0   (in place on agg)
__global__ void finish_T2(float* __restrict__ X2, const float* __restrict__ X0,
                          const float* __restrict__ dinv, int n) {
    int i = blockIdx.x * blockDim.x + threadIdx.x;
    if (i < n) X2[i] = -2.0f * dinv[i >> 6] * X2[i] - X0[i];
}

// ---------------- dense GEMM via V_WMMA_F32_16X16X4_F32 ----------------
// H[N x 64] = act([X0|X1|X2] (N x 192) @ W (192 x 64) + b)
// 256 threads = 8 waves; each wave: one 16-row M tile, 4 column tiles of 16.
// W staged in LDS K-pair-interleaved (Wp[pair][n][2], padded stride 160 dwords
// so lane-halves hit disjoint banks): B fragment = one conflict-free ds b64.
// A fragments are loaded straight from addrspace(1) pointers so they lower to
// global_load_b64 (LOADcnt only), decoupled from the LDS DScnt wait chain.
__global__ __launch_bounds__(256) void cheb_gemm_wmma(
        const float* __restrict__ X0,
        const float* __restrict__ X1,
        const float* __restrict__ X2,
        const float* __restrict__ W,   // [192][64] row-major
        const float* __restrict__ b,   // [64]
        float* __restrict__ out, int n_nodes, int do_relu) {
    __shared__ __align__(16) float Wp[NPAIR * PSTRIDE];  // 60 KB
    __shared__ float bl[D];
    for (int i = threadIdx.x; i < KD * D; i += blockDim.x) {
        int r = i >> 6;          // source row 0..191
        int n = i & 63;          // column
        Wp[(r >> 1) * PSTRIDE + n * 2 + (r & 1)] = W[i];
    }
    if (threadIdx.x < D) bl[threadIdx.x] = b[threadIdx.x];
    __syncthreads();

    const int wave = threadIdx.x >> 5;          // wave32
    const int lane = threadIdx.x & 31;
    const int tile = blockIdx.x * 8 + wave;     // 16-row tile
    const int n_tiles = (n_nodes + 15) >> 4;    // 6250 (exact)
    if (tile >= n_tiles) return;                // wave-uniform: EXEC stays all-1s

    const int row0 = tile << 4;
    const int h = lane >> 4;                    // lane half selects K {0,1} vs {2,3}
    const int l = lane & 15;

    v8f acc[4] = {v8f{}, v8f{}, v8f{}, v8f{}};

    // one Chebyshev-order block of the reduction; X must be a direct kernel-arg
    // pointer at each call site so the loads stay in the global address space
    auto kblock = [&](const float* __restrict__ X, int t) {
        const float* __restrict__ arow = X + (size_t)(row0 + l) * D;
#pragma unroll
        for (int k = 0; k < D; k += 4) {
            // A 16x4 f32 frag: VGPR0 = K=k+2h, VGPR1 = K=k+2h+1 for row (row0+l)
            v2f a = *reinterpret_cast<const v2f*>(arow + k + 2 * h);
            // B pair row: global pair index = (t*64 + k)/2 (+h for lane half)
            const v2f* __restrict__ bp = reinterpret_cast<const v2f*>(
                &Wp[(((t << 6) + k) >> 1) * PSTRIDE + h * PSTRIDE]);
#pragma unroll
            for (int nt = 0; nt < 4; ++nt) {
                v2f bf = bp[(nt << 4) + l];     // single aligned ds_load_b64
                acc[nt] = __builtin_amdgcn_wmma_f32_16x16x4_f32(
                    false, a, false, bf, (short)0, acc[nt], false, false);
            }
        }
    };
    kblock(X0, 0);
    kblock(X1, 1);
    kblock(X2, 2);

    // C/D 16x16 f32 layout: lane l -> col, VGPR v + 8h -> row
#pragma unroll
    for (int nt = 0; nt < 4; ++nt) {
        int n = (nt << 4) + l;
        float bias = bl[n];
#pragma unroll
        for (int v = 0; v < 8; ++v) {
            int m = row0 + v + 8 * h;
            float c = acc[nt][v] + bias;
            if (do_relu) c = fmaxf(c, 0.0f);
            out[(size_t)m * D + n] = c;
        }
    }
}

// ---------------- readout ----------------
__global__ void graph_accum(const float* __restrict__ H, const int* __restrict__ gid,
                            float* __restrict__ gsum, float* __restrict__ gcnt, int n_nodes) {
    int idx = blockIdx.x * blockDim.x + threadIdx.x;
    if (idx >= n_nodes * D) return;
    int node = idx >> 6;
    int f = idx & 63;
    int g = __ldg(gid + node);
    atomicAdd(&gsum[g * D + f], H[idx]);
    if (f == 0) atomicAdd(&gcnt[g], 1.0f);
}

__global__ void readout_mlp(const float* __restrict__ gsum, const float* __restrict__ gcnt,
                            const float* __restrict__ M0w, const float* __restrict__ M0b,
                            const float* __restrict__ M1w, const float* __restrict__ M1b,
                            const float* __restrict__ M2w, const float* __restrict__ M2b,
                            float* __restrict__ out) {
    __shared__ float hg[NGRAPH * 64];
    __shared__ float h1[NGRAPH * 32];
    __shared__ float h2[NGRAPH * 16];
    int t = threadIdx.x;
    for (int i = t; i < NGRAPH * 64; i += blockDim.x)
        hg[i] = gsum[i] / fmaxf(gcnt[i >> 6], 1.0f);
    __syncthreads();
    for (int i = t; i < NGRAPH * 32; i += blockDim.x) {
        int g = i >> 5, j = i & 31;
        float acc = M0b[j];
        for (int k = 0; k < 64; ++k) acc += hg[g * 64 + k] * M0w[k * 32 + j];
        h1[i] = fmaxf(acc, 0.0f);
    }
    __syncthreads();
    for (int i = t; i < NGRAPH * 16; i += blockDim.x) {
        int g = i >> 4, j = i & 15;
        float acc = M1b[j];
        for (int k = 0; k < 32; ++k) acc += h1[g * 32 + k] * M1w[k * 16 + j];
        h2[i] = fmaxf(acc, 0.0f);
    }
    __syncthreads();
    for (int i = t; i < NGRAPH * 3; i += blockDim.x) {
        int g = i / 3, j = i % 3;
        float acc = M2b[j];
        for (int k = 0; k < 16; ++k) acc += h2[g * 16 + k] * M2w[k * 3 + j];
        out[i] = acc;
    }
}

// ---------------- host orchestration ----------------
static inline int cdiv(int a, int b) { return (a + b - 1) / b; }

extern "C" void kernel_launch(void* const* d_in, const int* in_sizes, int n_in,
                              void* d_out, int out_size, void* d_ws, size_t ws_size,
                              hipStream_t stream) {
    const float* h   = (const float*)d_in[0];
    const int*   src = (const int*)d_in[1];
    const int*   dst = (const int*)d_in[2];
    const int*   gid = (const int*)d_in[3];
    const float* W0  = (const float*)d_in[4];
    const float* b0  = (const float*)d_in[5];
    const float* W1  = (const float*)d_in[6];
    const float* b1  = (const float*)d_in[7];
    const float* W2  = (const float*)d_in[8];
    const float* b2  = (const float*)d_in[9];
    const float* M0w = (const float*)d_in[10];
    const float* M0b = (const float*)d_in[11];
    const float* M1w = (const float*)d_in[12];
    const float* M1b = (const float*)d_in[13];
    const float* M2w = (const float*)d_in[14];
    const float* M2b = (const float*)d_in[15];

    const int N = in_sizes[0] / D;    // 100000
    const int E = in_sizes[1];        // 1600000
    const int N64 = N * D;

    float* ws   = (float*)d_ws;
    float* dinv = ws;                 // N
    float* X1   = dinv + N;           // N*64
    float* X2   = X1 + N64;           // N*64
    float* HA   = X2 + N64;           // N*64
    float* HB   = HA + N64;           // N*64
    float* gsum = HB + N64;           // 32*64
    float* gcnt = gsum + NGRAPH * D;  // 32

    const int TB = 256;

    // degree -> dinv
    zero_kernel<<<cdiv(N, TB), TB, 0, stream>>>(dinv, N);
    degree_kernel<<<cdiv(E, TB), TB, 0, stream>>>(dst, dinv, E);
    dinv_kernel<<<cdiv(N, TB), TB, 0, stream>>>(dinv, N);

    const int n_tiles = cdiv(N, 16);
    const int gemm_blocks = cdiv(n_tiles, 8);

    auto conv = [&](const float* Xin, float* Hout, const float* W, const float* b, int relu) {
        // T1 = -A_norm @ Xin
        zero_kernel<<<cdiv(N64, TB), TB, 0, stream>>>(X1, N64);
        spmm_scatter<<<cdiv(E * 16, TB), TB, 0, stream>>>(Xin, src, dst, dinv, X1, E);
        finish_T1<<<cdiv(N64, TB), TB, 0, stream>>>(X1, dinv, N64);
        // T2 = -2 * A_norm @ T1 - Xin
        zero_kernel<<<cdiv(N64, TB), TB, 0, stream>>>(X2, N64);
        spmm_scatter<<<cdiv(E * 16, TB), TB, 0, stream>>>(X1, src, dst, dinv, X2, E);
        finish_T2<<<cdiv(N64, TB), TB, 0, stream>>>(X2, Xin, dinv, N64);
        // Hout = act([Xin|T1|T2] @ W + b)
        cheb_gemm_wmma<<<gemm_blocks, TB, 0, stream>>>(Xin, X1, X2, W, b, Hout, N, relu);
    };

    conv(h,  HA, W0, b0, 1);
    conv(HA, HB, W1, b1, 1);
    conv(HB, HA, W2, b2, 0);

    // per-graph mean + MLP readout
    zero_kernel<<<cdiv(NGRAPH * D + NGRAPH, TB), TB, 0, stream>>>(gsum, NGRAPH * D + NGRAPH);
    graph_accum<<<cdiv(N64, TB), TB, 0, stream>>>(HA, gid, gsum, gcnt, N);
    readout_mlp<<<1, 256, 0, stream>>>(gsum, gcnt, M0w, M0b, M1w, M1b, M2w, M2b,
                                       (float*)d_out);
}